// MetaLearner_91147795956324
// MI455X (gfx1250) — compile-verified
//
#include <hip/hip_runtime.h>

#define HH 20
#define GG 80
#define EPSF 1e-5f
#define WAVES 4
#define NTHREADS (WAVES * 32)

typedef float v2f __attribute__((ext_vector_type(2)));
typedef float v8f __attribute__((ext_vector_type(8)));

// Parameter layout in LDS (float offsets)
enum {
  OFF_W1 = 0,
  OFF_B1   = OFF_W1 + HH,
  OFF_G1   = OFF_B1 + HH,
  OFF_BE1  = OFF_G1 + HH,
  OFF_WIH0 = OFF_BE1 + HH,        // 80*20
  OFF_GI0  = OFF_WIH0 + GG * HH,
  OFF_BI0  = OFF_GI0 + GG,
  OFF_BH0  = OFF_BI0 + GG,
  OFF_GO0  = OFF_BH0 + GG,
  OFF_BO0  = OFF_GO0 + HH,
  OFF_WIH1 = OFF_BO0 + HH,        // 80*20
  OFF_GI1  = OFF_WIH1 + GG * HH,
  OFF_BI1  = OFF_GI1 + GG,
  OFF_BH1  = OFF_BI1 + GG,
  OFF_GO1  = OFF_BH1 + GG,
  OFF_BO1  = OFF_GO1 + HH,
  OFF_WOUT = OFF_BO1 + HH,
  OFF_BOUT = OFF_WOUT + HH,
  NPARAM   = OFF_BOUT + 1
};

// Intra-wave LDS producer->consumer fence: same-wave LDS ops are in order,
// we only need completion + a compiler reordering barrier.
#define LDS_FENCE() asm volatile("s_wait_dscnt 0" ::: "memory")

// gfx1250 has a hardware V_TANH_F32 transcendental; use it if clang exposes it.
__device__ __forceinline__ float fast_tanh(float x) {
#if __has_builtin(__builtin_amdgcn_tanhf)
  return __builtin_amdgcn_tanhf(x);
#elif __has_builtin(__builtin_amdgcn_tanh_f32)
  return __builtin_amdgcn_tanh_f32(x);
#else
  return tanhf(x);
#endif
}

// sigmoid(x) == 0.5*tanh(x/2) + 0.5 (exact identity) -> rides v_tanh_f32
__device__ __forceinline__ float sigm(float v) {
  return fmaf(fast_tanh(v * 0.5f), 0.5f, 0.5f);
}

struct KParams {
  const float *x, *W1, *b1, *g1, *be1;
  const float *Wih0, *gi0, *bi0, *bh0, *go0, *bo0;
  const float *Wih1, *gi1, *bi1, *bh1, *go1, *bo1;
  const float *Wout, *bout;
  float* out;
  int n;
};

__global__ __launch_bounds__(NTHREADS) void lstm_wmma_kernel(KParams p) {
  __shared__ float sP[NPARAM];
  __shared__ float sAct[WAVES][HH][32];   // B tile: [K=20][N=32] per wave
  __shared__ float sGat[WAVES][GG][32];   // D tiles: [80 gates][32 samples] per wave

  const int tid = threadIdx.x;

  // ---- stage all (tiny) parameters into LDS once ----
  {
    auto cp = [&](const float* src, int off, int len) {
      for (int i = tid; i < len; i += NTHREADS) sP[off + i] = src[i];
    };
    cp(p.W1, OFF_W1, HH);   cp(p.b1, OFF_B1, HH);
    cp(p.g1, OFF_G1, HH);   cp(p.be1, OFF_BE1, HH);
    cp(p.Wih0, OFF_WIH0, GG * HH);
    cp(p.gi0, OFF_GI0, GG); cp(p.bi0, OFF_BI0, GG); cp(p.bh0, OFF_BH0, GG);
    cp(p.go0, OFF_GO0, HH); cp(p.bo0, OFF_BO0, HH);
    cp(p.Wih1, OFF_WIH1, GG * HH);
    cp(p.gi1, OFF_GI1, GG); cp(p.bi1, OFF_BI1, GG); cp(p.bh1, OFF_BH1, GG);
    cp(p.go1, OFF_GO1, HH); cp(p.bo1, OFF_BO1, HH);
    cp(p.Wout, OFF_WOUT, HH); cp(p.bout, OFF_BOUT, 1);
  }
  __syncthreads();

  const int lane = tid & 31;
  const int wv   = tid >> 5;
  const int half = lane >> 4;   // 0: lanes 0-15, 1: lanes 16-31
  const int l16  = lane & 15;

  const long long T = ((long long)p.n + 31) / 32;  // 32-sample tiles
  const long long stride = (long long)gridDim.x * WAVES;

  for (long long t = (long long)blockIdx.x * WAVES + wv; t < T; t += stride) {
    const int sidx = (int)(t * 32) + lane;
    const bool ok  = sidx < p.n;
    const float xv = ok ? p.x[sidx] : 0.0f;

    // ---- front layer: tanh(LN(x*W1 + b1)) ----  (one sample per lane)
    float s1 = 0.f;
#pragma unroll
    for (int j = 0; j < HH; ++j) s1 += fmaf(xv, sP[OFF_W1 + j], sP[OFF_B1 + j]);
    const float mu = s1 * (1.0f / HH);
    float s2 = 0.f;
#pragma unroll
    for (int j = 0; j < HH; ++j) {
      float d = fmaf(xv, sP[OFF_W1 + j], sP[OFF_B1 + j]) - mu;
      s2 = fmaf(d, d, s2);
    }
    const float inv = 1.0f / (sqrtf(s2 * (1.0f / (HH - 1))) + EPSF);
#pragma unroll
    for (int j = 0; j < HH; ++j) {
      float d = fmaf(xv, sP[OFF_W1 + j], sP[OFF_B1 + j]) - mu;
      sAct[wv][j][lane] = fast_tanh(fmaf(d * inv, sP[OFF_G1 + j], sP[OFF_BE1 + j]));
    }
    LDS_FENCE();

#pragma unroll
    for (int l = 0; l < 2; ++l) {
      // ---- per-layer A fragments from LDS (50 VGPRs, not 100) ----
      // 16x4 f32 A tile: VGPR0 = K (half? +2 : +0), VGPR1 = +1; lanes carry M.
      const float* W = sP + (l ? OFF_WIH1 : OFF_WIH0);
      v2f Af[5][5];
#pragma unroll
      for (int mt = 0; mt < 5; ++mt)
#pragma unroll
        for (int ks = 0; ks < 5; ++ks) {
          const float* wp = W + (mt * 16 + l16) * HH + ks * 4 + half * 2;
          v2f a; a[0] = wp[0]; a[1] = wp[1];
          Af[mt][ks] = a;
        }

      // ---- GEMM: gates[80][32] = Wih(80x20) @ act(20x32), fp32 WMMA ----
#pragma unroll
      for (int nt = 0; nt < 2; ++nt) {
        v2f Bf[5];
#pragma unroll
        for (int ks = 0; ks < 5; ++ks) {
          const int k = ks * 4 + half * 2;  // B 4x16 layout mirrors A halves
          v2f b;
          b[0] = sAct[wv][k][nt * 16 + l16];
          b[1] = sAct[wv][k + 1][nt * 16 + l16];
          Bf[ks] = b;
        }
#pragma unroll
        for (int mt = 0; mt < 5; ++mt) {
          v8f acc = {0.f, 0.f, 0.f, 0.f, 0.f, 0.f, 0.f, 0.f};
#pragma unroll
          for (int ks = 0; ks < 5; ++ks)
            acc = __builtin_amdgcn_wmma_f32_16x16x4_f32(
                false, Af[mt][ks], false, Bf[ks], (short)0, acc, false, false);
          // D layout: VGPR r -> row r (lanes 0-15) / row r+8 (lanes 16-31)
#pragma unroll
          for (int r = 0; r < 8; ++r)
            sGat[wv][mt * 16 + half * 8 + r][nt * 16 + l16] = acc[r];
        }
      }
      LDS_FENCE();

      // ---- pointwise LSTM cell, one sample per lane ----
      const int gio = l ? OFF_GI1 : OFF_GI0;
      const int bio = l ? OFF_BI1 : OFF_BI0;
      const int bho = l ? OFF_BH1 : OFF_BH0;   // LN(0,gh,bh) == bh
      const int goo = l ? OFF_GO1 : OFF_GO0;
      const int boo = l ? OFF_BO1 : OFF_BO0;

      // single-pass mean + variance over the 80 gate pre-activations
      float gs = 0.f, gq = 0.f;
#pragma unroll
      for (int g = 0; g < GG; ++g) {
        float y = sGat[wv][g][lane];
        gs += y;
        gq = fmaf(y, y, gq);
      }
      const float gmu  = gs * (1.0f / GG);
      const float gvar = (gq - (float)GG * gmu * gmu) * (1.0f / (GG - 1));
      const float ginv = 1.0f / (sqrtf(fmaxf(gvar, 0.0f)) + EPSF);

      float cx[HH], og[HH];
#pragma unroll
      for (int j = 0; j < HH; ++j) {
        float iv = fmaf((sGat[wv][j][lane] - gmu) * ginv, sP[gio + j], sP[bio + j]) + sP[bho + j];
        float ov = fmaf((sGat[wv][40 + j][lane] - gmu) * ginv, sP[gio + 40 + j], sP[bio + 40 + j]) + sP[bho + 40 + j];
        float gv = fmaf((sGat[wv][60 + j][lane] - gmu) * ginv, sP[gio + 60 + j], sP[bio + 60 + j]) + sP[bho + 60 + j];
        cx[j] = sigm(iv) * fast_tanh(gv);   // f*cx == 0 since cx_in == 0
        og[j] = sigm(ov);
      }
      float cs = 0.f;
#pragma unroll
      for (int j = 0; j < HH; ++j) cs += cx[j];
      const float cmu = cs * (1.0f / HH);
      float cq = 0.f;
#pragma unroll
      for (int j = 0; j < HH; ++j) {
        float d = cx[j] - cmu;
        cq = fmaf(d, d, cq);
      }
      const float cinv = 1.0f / (sqrtf(cq * (1.0f / (HH - 1))) + EPSF);

      if (l == 0) {
#pragma unroll
        for (int j = 0; j < HH; ++j)
          sAct[wv][j][lane] = og[j] * fast_tanh(fmaf((cx[j] - cmu) * cinv, sP[goo + j], sP[boo + j]));
        LDS_FENCE();
      } else {
        float o = sP[OFF_BOUT];
#pragma unroll
        for (int j = 0; j < HH; ++j) {
          float h = og[j] * fast_tanh(fmaf((cx[j] - cmu) * cinv, sP[goo + j], sP[boo + j]));
          o = fmaf(h, sP[OFF_WOUT + j], o);
        }
        if (ok) p.out[sidx] = o;
      }
    }
  }
}

extern "C" void kernel_launch(void* const* d_in, const int* in_sizes, int n_in,
                              void* d_out, int out_size, void* d_ws, size_t ws_size,
                              hipStream_t stream) {
  (void)n_in; (void)out_size; (void)d_ws; (void)ws_size;
  KParams p;
  p.x    = (const float*)d_in[0];
  p.W1   = (const float*)d_in[1];
  p.b1   = (const float*)d_in[2];
  p.g1   = (const float*)d_in[3];
  p.be1  = (const float*)d_in[4];
  p.Wih0 = (const float*)d_in[5];
  // d_in[6] = Whh0 unused (hx == 0)
  p.gi0  = (const float*)d_in[7];
  p.bi0  = (const float*)d_in[8];
  // d_in[9] = gh0 unused (LN(0,gh,bh) == bh)
  p.bh0  = (const float*)d_in[10];
  p.go0  = (const float*)d_in[11];
  p.bo0  = (const float*)d_in[12];
  p.Wih1 = (const float*)d_in[13];
  // d_in[14] = Whh1 unused
  p.gi1  = (const float*)d_in[15];
  p.bi1  = (const float*)d_in[16];
  // d_in[17] = gh1 unused
  p.bh1  = (const float*)d_in[18];
  p.go1  = (const float*)d_in[19];
  p.bo1  = (const float*)d_in[20];
  p.Wout = (const float*)d_in[21];
  p.bout = (const float*)d_in[22];
  p.out  = (float*)d_out;
  p.n    = in_sizes[0];

  long long T = ((long long)p.n + 31) / 32;
  long long blocks = (T + WAVES - 1) / WAVES;
  if (blocks > 3072) blocks = 3072;
  if (blocks < 1) blocks = 1;
  lstm_wmma_kernel<<<(int)blocks, NTHREADS, 0, stream>>>(p);
}